// Malis_89704686944453
// MI455X (gfx1250) — compile-verified
//
#include <hip/hip_runtime.h>
#include <hip/hip_bf16.h>
#include <stdint.h>

// Problem geometry (fixed by the reference setup).
#define HH 128
#define WW 128
#define VV (HH * WW)      // 16384 vertices per image
#define EE (2 * VV)       // 32768 edges per image
#define NIMG 4
#define NLBL 64
#define NTHREADS 1024
#define SMEM_BYTES (320 * 1024)

typedef __attribute__((ext_vector_type(2))) float v2f;
typedef __attribute__((ext_vector_type(8))) float v8f;
typedef __attribute__((ext_vector_type(4))) unsigned int v4u;
typedef __attribute__((ext_vector_type(8))) int v8i;
typedef __attribute__((ext_vector_type(4))) int v4i;

#if __has_builtin(__builtin_amdgcn_tensor_load_to_lds)
#define HAVE_TDM 1
#else
#define HAVE_TDM 0
#endif

#if __has_builtin(__builtin_amdgcn_wmma_f32_16x16x4_f32)
#define HAVE_WMMA4 1
#else
#define HAVE_WMMA4 0
#endif

__device__ __forceinline__ float waveSum32(float x) {
#pragma unroll
  for (int off = 16; off > 0; off >>= 1) x += __shfl_xor(x, off, 32);
  return x;
}

extern "C" __global__ __launch_bounds__(NTHREADS)
void malis_kernel(const float* __restrict__ aff, const int* __restrict__ gt,
                  float* __restrict__ out, uint32_t* __restrict__ cnt) {
  extern __shared__ unsigned char smem[];
  const int n = blockIdx.x;
  const int tid = (int)threadIdx.x;

  // LDS layout (time-multiplexed):
  //  phase 1-2: keys u64[32768]            @ [0, 256K)
  //             staged aff f32[32768]      @ [128K, 256K)  (inside key region)
  //  phase 3+ : edges u16[32768]           @ [256K, 320K)
  //             parent u32[16384]          @ [0, 64K)
  //             size   u32[16384]          @ [64K, 128K)
  //             sums   f32[16384]          @ [128K, 192K)
  unsigned long long* keys = (unsigned long long*)smem;
  float* affLds = (float*)(smem + 131072);
  uint16_t* edges = (uint16_t*)(smem + 262144);
  uint32_t* parent = (uint32_t*)smem;
  uint32_t* szArr = (uint32_t*)(smem + 65536);
  float* sums = (float*)(smem + 131072);

  const float* affImg = aff + (size_t)n * EE;
  uint32_t* cntImg = cnt + ((size_t)n << 20);  // 16384 * 64 = 2^20 u32 per image

  // ---- Phase 0: zero output slices for this image; init one-hot counts (L2-resident).
  for (int i = tid; i < EE; i += NTHREADS) {
    out[(size_t)n * EE + i] = 0.0f;
    out[(size_t)(NIMG + n) * EE + i] = 0.0f;
  }
  for (int v = tid; v < VV; v += NTHREADS) {
    int lbl = gt[(size_t)n * VV + v];
    uint32_t* c = cntImg + ((size_t)v << 6);
#pragma unroll
    for (int l = 0; l < NLBL; ++l) c[l] = 0u;
    if (lbl > 0 && lbl < NLBL) c[lbl] = 1u;
  }

  // ---- Phase 1: stage the 128KB affinity slab into LDS via the Tensor Data Mover.
#if HAVE_TDM
  if ((tid >> 5) == 0) {  // wave 0 only issues the DMA
    uint64_t ga = (uint64_t)(uintptr_t)affImg;
    uint32_t la = (uint32_t)(uintptr_t)affLds;
    v4u g0;
    g0.x = 1u;                                   // count=1 valid descriptor
    g0.y = la;                                   // lds_addr
    g0.z = (uint32_t)ga;                         // global_addr[31:0]
    g0.w = (uint32_t)((ga >> 32) & 0x01FFFFFFull) | (2u << 30);  // addr[56:32] | type=2
    v8i g1;
    g1[0] = 0x00020000;        // data_size = 4 bytes; no multicast/pad/iterate
    g1[1] = (int)0x80000000;   // tensor_dim0[15:0] = 32768 in bits [63:48]
    g1[2] = 0x00010000;        // tensor_dim0[31:16]=0 ; tensor_dim1 = 1
    g1[3] = (int)0x80000000;   // tensor_dim1 hi = 0 ; tile_dim0 = 32768
    g1[4] = 0x00000001;        // tile_dim1 = 1, tile_dim2 = 0
    g1[5] = 32768;             // tensor_dim0_stride (elements)
    g1[6] = 0;
    g1[7] = 0;
    v4i g2 = {0, 0, 0, 0};
    v4i g3 = {0, 0, 0, 0};
    v8i g4 = {0, 0, 0, 0, 0, 0, 0, 0};  // 6-arg variant: extra descriptor group
    __builtin_amdgcn_tensor_load_to_lds(g0, g1, g2, g3, g4, 0);
#if __has_builtin(__builtin_amdgcn_s_wait_tensorcnt)
    __builtin_amdgcn_s_wait_tensorcnt(0);
#else
    asm volatile("s_wait_tensorcnt 0x0" ::: "memory");
#endif
  }
  __syncthreads();
#else
  for (int i = tid; i < EE; i += NTHREADS) affLds[i] = affImg[i];
  __syncthreads();
#endif

  // ---- Phase 1b: expand staged f32 affinities into 64-bit stable sort keys, in place.
  // Ascending chunks with read->barrier->write: write at byte 8i only clobbers staged
  // element 2i-32768 < i, which has already been consumed.
  for (int c = 0; c < EE / NTHREADS; ++c) {
    int i = c * NTHREADS + tid;
    float f = affLds[i];
    uint32_t u = __float_as_uint(f);
    uint32_t mk = (u & 0x80000000u) ? ~u : (u | 0x80000000u);  // order-preserving map
    unsigned long long comp =
        ((unsigned long long)(~mk) << 32) | (unsigned)i;  // ascending == stable argsort(-w)
    __syncthreads();
    keys[i] = comp;
    __syncthreads();
  }

  // ---- Phase 2: bitonic sort (ascending) of 32768 keys entirely in LDS.
  for (unsigned k = 2; k <= (unsigned)EE; k <<= 1) {
    for (unsigned j = k >> 1; j > 0; j >>= 1) {
      for (unsigned t = (unsigned)tid; t < (unsigned)(EE / 2); t += NTHREADS) {
        unsigned p = ((t & ~(j - 1)) << 1) | (t & (j - 1));
        unsigned q = p | j;
        bool up = ((p & k) == 0);
        unsigned long long a = keys[p];
        unsigned long long b = keys[q];
        if ((a > b) == up) {
          keys[p] = b;
          keys[q] = a;
        }
      }
      __syncthreads();
    }
  }

  // ---- Phase 3: compact sorted edge indices to u16 (separate 64KB region).
  for (int i = tid; i < EE; i += NTHREADS) edges[i] = (uint16_t)(keys[i] & 0xFFFFu);
  __syncthreads();

  // ---- Phase 4: init union-find state over the retired key region.
  for (int v = tid; v < VV; v += NTHREADS) {
    parent[v] = (uint32_t)v;
    szArr[v] = 1u;
    int lbl = gt[(size_t)n * VV + v];
    sums[v] = (lbl > 0 && lbl < NLBL) ? 1.0f : 0.0f;
  }
  __threadfence();  // make count-table init + output zeros globally visible before wave 0 writes
  __syncthreads();

  // ---- Phase 5: serial Kruskal scan by wave 0; labels spread 2-per-lane across the wave.
  if (tid >= 32) return;
  const int lane = tid;
  // Chunk index mapping for the 16x16x4 WMMA A/B layouts:
  // lanes 0-15 hold elements {4L, 4L+1}; lanes 16-31 hold {4(L-16)+2, 4(L-16)+3}.
  const int base = (lane < 16) ? (lane << 2) : (((lane - 16) << 2) | 2);

  for (int e = 0; e < EE; ++e) {
    int idx = (int)edges[e];
    int uu, vv;
    if (idx < VV) {
      uu = idx;
      vv = (uu < VV - WW) ? (uu + WW) : uu;  // channel 0: (y,x)-(y+1,x)
    } else {
      int t2 = idx - VV;
      uu = t2;
      vv = ((t2 & (WW - 1)) < WW - 1) ? (t2 + 1) : t2;  // channel 1: (y,x)-(y,x+1)
    }
    if (uu == vv) continue;  // boundary self-edge

    int ru = uu;
    for (;;) {  // find with path halving (all lanes identical -> LDS broadcast)
      uint32_t p = parent[ru];
      if (p == (uint32_t)ru) break;
      uint32_t gp = parent[p];
      parent[ru] = gp;
      ru = (int)gp;
    }
    int rv = vv;
    for (;;) {
      uint32_t p = parent[rv];
      if (p == (uint32_t)rv) break;
      uint32_t gp = parent[p];
      parent[rv] = gp;
      rv = (int)gp;
    }
    if (ru == rv) continue;

    uint32_t* cA = cntImg + ((size_t)ru << 6);
    uint32_t* cB = cntImg + ((size_t)rv << 6);
    uint32_t ia0 = cA[base], ia1 = cA[base + 1];
    uint32_t ib0 = cB[base], ib1 = cB[base + 1];
    float a0 = (float)ia0, a1 = (float)ia1;
    float b0 = (float)ib0, b1 = (float)ib1;

    float pos;
#if HAVE_WMMA4
    // D = A(16x4) x B(4x16): D[i][i] = dot of chunk i of cntA with chunk i of cntB,
    // so trace(D) is the full 64-element dot product.
    {
      v2f A = {a0, a1};
      v2f B = {b0, b1};
      v8f Cz = {0.f, 0.f, 0.f, 0.f, 0.f, 0.f, 0.f, 0.f};
      v8f D = __builtin_amdgcn_wmma_f32_16x16x4_f32(false, A, false, B, (short)0, Cz,
                                                    false, false);
      // Diagonal D[i][i]: lanes 0-7 element i=lane (vgpr=lane); lanes 24-31 element
      // i=lane-16 (vgpr=lane-24). Other lanes contribute zero.
      int s = (lane < 8) ? lane : (lane - 24);
      float diag = (lane < 8 || lane >= 24) ? D[s & 7] : 0.0f;
      pos = waveSum32(diag);
    }
#else
    pos = waveSum32(a0 * b0 + a1 * b1);
#endif

    float sA = sums[ru], sB = sums[rv];
    float neg = sA * sB - pos;

    uint32_t szu = szArr[ru], szv = szArr[rv];
    int big = (szu >= szv) ? ru : rv;     // union by size (ties -> ru, as reference)
    int small = ru + rv - big;

    uint32_t* cBig = cntImg + ((size_t)big << 6);
    uint32_t* cSml = cntImg + ((size_t)small << 6);
    cBig[base] = ia0 + ib0;
    cBig[base + 1] = ia1 + ib1;
    cSml[base] = 0u;
    cSml[base + 1] = 0u;

    if (lane == 0) {
      parent[small] = (uint32_t)big;
      szArr[big] = szu + szv;
      sums[big] = sA + sB;
      sums[small] = 0.0f;
      float f = affImg[idx];
      out[(size_t)n * EE + idx] = -(1.0f - f) * pos;   // pos gradient
      out[(size_t)(NIMG + n) * EE + idx] = f * neg;    // neg gradient
    }
  }
}

extern "C" void kernel_launch(void* const* d_in, const int* in_sizes, int n_in,
                              void* d_out, int out_size, void* d_ws, size_t ws_size,
                              hipStream_t stream) {
  const float* aff = (const float*)d_in[0];      // [4, 2, 128, 128] f32
  const int* gt = (const int*)d_in[1];           // [4, 1, 128, 128] int labels
  float* out = (float*)d_out;                    // [2, 4, 2, 128, 128] f32
  uint32_t* cnt = (uint32_t*)d_ws;               // 4 * 16384 * 64 u32 = 16 MB

  (void)in_sizes; (void)n_in; (void)out_size; (void)ws_size;

  // Allow the full 320KB/WGP dynamic LDS allocation.
  (void)hipFuncSetAttribute((const void*)malis_kernel,
                            hipFuncAttributeMaxDynamicSharedMemorySize, SMEM_BYTES);

  malis_kernel<<<NIMG, NTHREADS, SMEM_BYTES, stream>>>(aff, gt, out, cnt);
}